// GNNModel_31172872634884
// MI455X (gfx1250) — compile-verified
//
#include <hip/hip_runtime.h>
#include <stdint.h>

typedef __attribute__((ext_vector_type(2))) float v2f;
typedef __attribute__((ext_vector_type(8))) float v8f;

#define N_NODES  100000
#define N_EDGES  3200000
#define N_GRAPHS 256
#define D_IN     320
#define D_H      64

// ---------------------------------------------------------------- TDM support
#if defined(__has_builtin)
#  if __has_builtin(__builtin_amdgcn_tensor_load_to_lds) && \
      __has_builtin(__builtin_amdgcn_s_wait_tensorcnt)
#    define USE_TDM 1
#  endif
#endif
#ifndef USE_TDM
#  define USE_TDM 0
#endif

#if USE_TDM
typedef unsigned int tdm_v4u __attribute__((ext_vector_type(4)));
typedef int          tdm_v8i __attribute__((ext_vector_type(8)));
typedef int          tdm_v4i __attribute__((ext_vector_type(4)));

// 2D tile DMA: Global (row-major, 4-byte elems, stride0 elems/row) -> LDS.
// Descriptor packing per CDNA5 ISA "Tensor DMA Descriptor (D#)" groups 0/1.
__device__ __forceinline__ void tdm_load_2d_to_lds(
    const void* gaddr, unsigned lds_byte_addr,
    unsigned tensor_d0, unsigned tensor_d1,
    unsigned tile_d0, unsigned tile_d1, unsigned long long stride0) {
    unsigned long long ga = (unsigned long long)(uintptr_t)gaddr;
    tdm_v4u g0;
    g0[0] = 1u;                                        // count=1 (valid user D#)
    g0[1] = lds_byte_addr;                             // lds_addr[31:0]
    g0[2] = (unsigned)(ga & 0xFFFFFFFFull);            // global_addr[31:0]
    g0[3] = (unsigned)((ga >> 32) & 0x1FFFFFFull)      // global_addr[56:32]
          | (2u << 30);                                // type = 2 ("image")
    tdm_v8i g1;
    g1[0] = (int)(2u << 16);                           // data_size=2 (4B), wg_mask=0
    g1[1] = (int)((tensor_d0 & 0xFFFFu) << 16);        // tensor_dim0[15:0] @ bits 63:48
    g1[2] = (int)(((tensor_d0 >> 16) & 0xFFFFu)        // tensor_dim0[31:16]
          | ((tensor_d1 & 0xFFFFu) << 16));            // tensor_dim1[15:0]
    g1[3] = (int)(((tensor_d1 >> 16) & 0xFFFFu)        // tensor_dim1[31:16]
          | ((tile_d0 & 0xFFFFu) << 16));              // tile_dim0
    g1[4] = (int)(tile_d1 & 0xFFFFu);                  // tile_dim1 (tile_dim2 = 0)
    g1[5] = (int)(unsigned)(stride0 & 0xFFFFFFFFull);  // tensor_dim0_stride[31:0]
    g1[6] = (int)(unsigned)((stride0 >> 32) & 0xFFFFull); // stride[47:32]; dim1_stride=0
    g1[7] = 0;
    tdm_v4i gz = {0, 0, 0, 0};                         // groups 2/3: tile_dim3/4 = 0
#if __clang_major__ >= 23
    tdm_v8i gz8 = {0, 0, 0, 0, 0, 0, 0, 0};
    __builtin_amdgcn_tensor_load_to_lds(g0, g1, gz, gz, gz8, 0);
#else
    __builtin_amdgcn_tensor_load_to_lds(g0, g1, gz, gz, 0);
#endif
}
#endif // USE_TDM

// ---------------------------------------------------------------- norm setup
__global__ void deg_init_k(float* __restrict__ deg) {
    int i = blockIdx.x * 256 + threadIdx.x;
    if (i < N_NODES) deg[i] = 1.0f;            // self-loop contribution
}

__global__ void deg_count_k(const int* __restrict__ dst, float* __restrict__ deg) {
    int e = blockIdx.x * 256 + threadIdx.x;
    if (e < N_EDGES) atomicAdd(&deg[dst[e]], 1.0f);
}

__global__ void dinv_k(const float* __restrict__ deg, float* __restrict__ dinv) {
    int i = blockIdx.x * 256 + threadIdx.x;
    if (i < N_NODES) {
        float d = deg[i];
        dinv[i] = (d > 0.0f) ? rsqrtf(d) : 0.0f;
    }
}

__global__ void norm_k(const int* __restrict__ src, const int* __restrict__ dst,
                       const float* __restrict__ dinv, float* __restrict__ normE) {
    int e = blockIdx.x * 256 + threadIdx.x;
    if (e < N_EDGES) normE[e] = dinv[src[e]] * dinv[dst[e]];
}

// ------------------------------------------------- fp32 WMMA GEMM  T = H @ W
// One wave -> 16 rows x 64 cols.  Block = 2 waves (32 rows).  W is staged into
// LDS in 64x64 chunks by the Tensor Data Mover (wave 0 issues, both waves sync
// at the workgroup barrier).  Also writes agg initialized with the self-loop
// term T[i]*dinv[i]^2.
__global__ __launch_bounds__(64) void gemm_selfloop_k(
    const float* __restrict__ Hin, const float* __restrict__ W,
    const float* __restrict__ dinv, float* __restrict__ T,
    float* __restrict__ agg, int Din) {
    __shared__ float sW[64 * D_H];                 // 16 KB chunk of W

    const int wave    = threadIdx.x >> 5;
    const int lane    = threadIdx.x & 31;
    const int half    = lane >> 4;                 // 0: K=0,1  1: K=2,3
    const int l16     = lane & 15;
    const int rowtile = blockIdx.x * 2 + wave;
    const int row     = rowtile * 16 + l16;        // A-matrix row for this lane

    v8f acc0 = {}, acc1 = {}, acc2 = {}, acc3 = {};

    for (int kc = 0; kc < Din; kc += 64) {
        __syncthreads();                           // WAR: previous chunk's reads done
#if USE_TDM
        if (threadIdx.x < 32) {                    // wave 0 drives the DMA engine
            tdm_load_2d_to_lds(W + (size_t)kc * D_H,
                               (unsigned)(uintptr_t)sW,   // low 32 bits = LDS offset
                               /*tensor_d0=*/D_H, /*tensor_d1=*/(unsigned)(Din - kc),
                               /*tile_d0=*/64, /*tile_d1=*/64, /*stride0=*/D_H);
            __builtin_amdgcn_s_wait_tensorcnt(0);
        }
        __syncthreads();                           // publish LDS tile to wave 1
#else
        { // fallback: cooperative stage of W[kc:kc+64][0:64] -> LDS (float4)
            const float4* gw  = (const float4*)(W + (size_t)kc * D_H);
            float4*       sw4 = (float4*)sW;
            for (int i = threadIdx.x; i < (64 * D_H) / 4; i += 64) sw4[i] = gw[i];
        }
        __syncthreads();
#endif

        for (int k0 = 0; k0 < 64; k0 += 4) {
            const int ka = k0 + 2 * half;          // this lane's K within the step
            v2f a = *(const v2f*)(Hin + (size_t)row * Din + kc + ka);

            const float* wr = sW + ka * D_H + l16; // B: rows ka/ka+1, col = ntile*16+l16
            v2f b0, b1, b2, b3;
            b0.x = wr[0];   b0.y = wr[D_H + 0];
            b1.x = wr[16];  b1.y = wr[D_H + 16];
            b2.x = wr[32];  b2.y = wr[D_H + 32];
            b3.x = wr[48];  b3.y = wr[D_H + 48];

            acc0 = __builtin_amdgcn_wmma_f32_16x16x4_f32(false, a, false, b0, (short)0, acc0, false, false);
            acc1 = __builtin_amdgcn_wmma_f32_16x16x4_f32(false, a, false, b1, (short)0, acc1, false, false);
            acc2 = __builtin_amdgcn_wmma_f32_16x16x4_f32(false, a, false, b2, (short)0, acc2, false, false);
            acc3 = __builtin_amdgcn_wmma_f32_16x16x4_f32(false, a, false, b3, (short)0, acc3, false, false);
        }
    }

    // D layout: VGPR r holds row (r + 8*half), N = l16 within each 16-col tile.
#pragma unroll
    for (int r = 0; r < 8; ++r) {
        int   orow = rowtile * 16 + r + 8 * half;
        float dv   = dinv[orow];
        float dv2  = dv * dv;
        float* trow = T   + (size_t)orow * D_H + l16;
        float* arow = agg + (size_t)orow * D_H + l16;
        float v0 = acc0[r], v1 = acc1[r], v2 = acc2[r], v3 = acc3[r];
        trow[0] = v0;  trow[16] = v1;  trow[32] = v2;  trow[48] = v3;
        arow[0] = v0 * dv2;  arow[16] = v1 * dv2;  arow[32] = v2 * dv2;  arow[48] = v3 * dv2;
    }
}

// ------------------------------------------------- edge scatter (L2 atomics)
__global__ void scatter_k(const int* __restrict__ src, const int* __restrict__ dst,
                          const float* __restrict__ normE, const float* __restrict__ T,
                          float* __restrict__ agg) {
    long long id = (long long)blockIdx.x * 256 + threadIdx.x;
    if (id >= (long long)N_EDGES * D_H) return;
    int e = (int)(id >> 6);
    int c = (int)(id & 63);
    int s = src[e], d = dst[e];
    float v = T[(size_t)s * D_H + c] * normE[e];
    atomicAdd(&agg[(size_t)d * D_H + c], v);
}

// ------------------------------------------------- bias + optional ReLU
__global__ void bias_act_k(const float* __restrict__ agg, const float* __restrict__ b,
                           float* __restrict__ Hout, int do_relu) {
    int id = blockIdx.x * 256 + threadIdx.x;
    if (id >= N_NODES * D_H) return;
    float v = agg[id] + b[id & 63];
    Hout[id] = do_relu ? fmaxf(v, 0.0f) : v;
}

// ------------------------------------------------- mean pool + linear head
__global__ void zero_pool_k(float* __restrict__ sums, float* __restrict__ cnts) {
    int i = blockIdx.x * 256 + threadIdx.x;
    if (i < N_GRAPHS * D_H) sums[i] = 0.0f;
    if (i < N_GRAPHS) cnts[i] = 0.0f;
}

__global__ void pool_k(const float* __restrict__ H3, const int* __restrict__ batch,
                       float* __restrict__ sums, float* __restrict__ cnts) {
    int id = blockIdx.x * 256 + threadIdx.x;
    if (id >= N_NODES * D_H) return;
    int i = id >> 6, c = id & 63;
    int g = batch[i];
    atomicAdd(&sums[g * D_H + c], H3[id]);
    if (c == 0) atomicAdd(&cnts[g], 1.0f);
}

__global__ void head_k(const float* __restrict__ sums, const float* __restrict__ cnts,
                       const float* __restrict__ Wl, const float* __restrict__ bl,
                       float* __restrict__ out) {
    int g = threadIdx.x;                           // 256 graphs, 1 block
    float inv = 1.0f / fmaxf(cnts[g], 1.0f);
    float acc = bl[0];
#pragma unroll
    for (int c = 0; c < D_H; ++c) acc += sums[g * D_H + c] * inv * Wl[c];
    out[g] = acc;
}

// ---------------------------------------------------------------- launcher
extern "C" void kernel_launch(void* const* d_in, const int* in_sizes, int n_in,
                              void* d_out, int out_size, void* d_ws, size_t ws_size,
                              hipStream_t stream) {
    const float* x  = (const float*)d_in[0];
    const int*   ei = (const int*)d_in[1];
    const int    E  = in_sizes[1] / 2;
    const int* srcI = ei;
    const int* dstI = ei + E;
    const int*   batch = (const int*)d_in[2];
    const float* W1 = (const float*)d_in[3];
    const float* b1 = (const float*)d_in[4];
    const float* W2 = (const float*)d_in[5];
    const float* b2 = (const float*)d_in[6];
    const float* W3 = (const float*)d_in[7];
    const float* b3 = (const float*)d_in[8];
    const float* Wl = (const float*)d_in[9];
    const float* bl = (const float*)d_in[10];
    float* out = (float*)d_out;

    // workspace carve-out (256B aligned), ~91 MB total
    char*  ws  = (char*)d_ws;
    size_t off = 0;
    auto carve = [&](size_t bytes) {
        void* p = ws + off;
        off += (bytes + 255) & ~(size_t)255;
        return p;
    };
    float* deg   = (float*)carve((size_t)N_NODES * 4);
    float* dinv  = (float*)carve((size_t)N_NODES * 4);
    float* normE = (float*)carve((size_t)N_EDGES * 4);
    float* T     = (float*)carve((size_t)N_NODES * D_H * 4);
    float* agg   = (float*)carve((size_t)N_NODES * D_H * 4);
    float* H     = (float*)carve((size_t)N_NODES * D_H * 4);
    float* sums  = (float*)carve((size_t)N_GRAPHS * D_H * 4);
    float* cnts  = (float*)carve((size_t)N_GRAPHS * 4);
    (void)ws_size; (void)n_in; (void)out_size;

    const int nodeBlk  = (N_NODES + 255) / 256;
    const int edgeBlk  = (N_EDGES + 255) / 256;
    const int nhBlk    = (N_NODES * D_H + 255) / 256;
    const int ehBlk    = (int)(((long long)N_EDGES * D_H + 255) / 256);
    const int gemmBlk  = N_NODES / 32;             // 2 row-tiles per block

    // ---- normalization (once, reused by all 3 layers) ----
    deg_init_k<<<nodeBlk, 256, 0, stream>>>(deg);
    zero_pool_k<<<(N_GRAPHS * D_H + 255) / 256, 256, 0, stream>>>(sums, cnts);
    deg_count_k<<<edgeBlk, 256, 0, stream>>>(dstI, deg);
    dinv_k<<<nodeBlk, 256, 0, stream>>>(deg, dinv);
    norm_k<<<edgeBlk, 256, 0, stream>>>(srcI, dstI, dinv, normE);

    // ---- layer 1: x[100000,320] @ W1 -> relu ----
    gemm_selfloop_k<<<gemmBlk, 64, 0, stream>>>(x, W1, dinv, T, agg, D_IN);
    scatter_k<<<ehBlk, 256, 0, stream>>>(srcI, dstI, normE, T, agg);
    bias_act_k<<<nhBlk, 256, 0, stream>>>(agg, b1, H, 1);

    // ---- layer 2 ----
    gemm_selfloop_k<<<gemmBlk, 64, 0, stream>>>(H, W2, dinv, T, agg, D_H);
    scatter_k<<<ehBlk, 256, 0, stream>>>(srcI, dstI, normE, T, agg);
    bias_act_k<<<nhBlk, 256, 0, stream>>>(agg, b2, H, 1);

    // ---- layer 3 (no relu) ----
    gemm_selfloop_k<<<gemmBlk, 64, 0, stream>>>(H, W3, dinv, T, agg, D_H);
    scatter_k<<<ehBlk, 256, 0, stream>>>(srcI, dstI, normE, T, agg);
    bias_act_k<<<nhBlk, 256, 0, stream>>>(agg, b3, H, 0);

    // ---- mean pool + head ----
    pool_k<<<nhBlk, 256, 0, stream>>>(H, batch, sums, cnts);
    head_k<<<1, 256, 0, stream>>>(sums, cnts, Wl, bl, out);
}